// Wide_Transformer_21208548507831
// MI455X (gfx1250) — compile-verified
//
#include <hip/hip_runtime.h>
#include <hip/hip_bf16.h>

// ---------------- problem constants ----------------
#define B4    4
#define CIN   48          // dim
#define DIM1  384         // dim1 = 48*8
#define C4    1536        // dim1 * 4
#define HEADS 8
#define HC    48          // channels per head
#define HW    16384       // 128*128
#define IMG   128

typedef __attribute__((ext_vector_type(2))) float v2f;
typedef __attribute__((ext_vector_type(8))) float v8f;
typedef __attribute__((ext_vector_type(4))) int   v4i;

#define AS1 __attribute__((address_space(1)))
#define AS3 __attribute__((address_space(3)))

#if __has_builtin(__builtin_amdgcn_global_load_async_to_lds_b128) && \
    __has_builtin(__builtin_amdgcn_s_wait_asynccnt)
#define HAVE_ASYNC_LDS 1
#endif

// =====================================================================
// K1: qkv 1x1 conv: qkv[b,o,p] = sum_c W[o,c] * x[b,c,p]
// per-batch GEMM (1536 x 48) x (48 x 16384), fp32 WMMA 16x16x4
// grid (32, 96, 4), block 256 (8 waves); each wave: 16x64 output
// =====================================================================
__global__ __launch_bounds__(256) void k1_qkv(const float* __restrict__ x,
                                              const float* __restrict__ w,
                                              float* __restrict__ qkv) {
  const int b   = blockIdx.z;
  const int m0  = blockIdx.y * 16;
  const int tid = threadIdx.x;
  const int wave = tid >> 5, lane = tid & 31;
  __shared__ float As[16 * 49];
  for (int i = tid; i < 16 * 48; i += 256) {
    int r = i / 48, c = i % 48;
    As[r * 49 + c] = w[(m0 + r) * 48 + c];
  }
  __syncthreads();
  const float* xb = x + (size_t)b * CIN * HW;
  float* ob = qkv + (size_t)b * C4 * HW + (size_t)m0 * HW;
  const int lr = lane & 15;   // M (for A) / N (for B,C) within half
  const int hi = lane >> 4;   // half select: K pair 0/1 vs 2/3
  for (int nt = 0; nt < 4; ++nt) {
    const int p0 = blockIdx.x * 512 + (wave * 4 + nt) * 16;
    v8f acc = {};
    for (int kk = 0; kk < 48; kk += 4) {
      const int ka = kk + hi * 2;
      v2f a, bb;
      a.x  = As[lr * 49 + ka];
      a.y  = As[lr * 49 + ka + 1];
      bb.x = xb[(size_t)ka * HW + p0 + lr];
      bb.y = xb[(size_t)(ka + 1) * HW + p0 + lr];
      acc = __builtin_amdgcn_wmma_f32_16x16x4_f32(false, a, false, bb,
                                                  (short)0, acc, false, false);
    }
#pragma unroll
    for (int r = 0; r < 8; ++r)
      ob[(size_t)(r + 8 * hi) * HW + p0 + lr] = acc[r];
  }
}

// =====================================================================
// K2: grouped 3x3 conv, 192 groups, 8 in / 8 out channels per group.
// Blocks in the "l" quarter (g >= 144) also accumulate spatial sums
// for channel attention. grid (64, 192, 4), block 256 (16x16 tile).
// =====================================================================
__global__ __launch_bounds__(256) void k2_gconv(const float* __restrict__ qkv,
                                                const float* __restrict__ dw,
                                                float* __restrict__ out,
                                                float* __restrict__ lsum) {
  const int b = blockIdx.z, g = blockIdx.y;
  const int ty0 = (blockIdx.x >> 3) * 16, tx0 = (blockIdx.x & 7) * 16;
  const int tid = threadIdx.x;
  __shared__ float ws_[8 * 72];          // [oc][ic][3x3]
  __shared__ float in_s[8][18][18];      // haloed input tile
  __shared__ float red[256];
  for (int i = tid; i < 576; i += 256) ws_[i] = dw[(size_t)g * 576 + i];
  const float* qb = qkv + ((size_t)b * C4 + g * 8) * HW;
  for (int i = tid; i < 8 * 18 * 18; i += 256) {
    int ic = i / 324, rem = i % 324, yy = rem / 18, xx = rem % 18;
    int gy = ty0 + yy - 1, gx = tx0 + xx - 1;
    float v = 0.f;
    if (gy >= 0 && gy < IMG && gx >= 0 && gx < IMG)
      v = qb[(size_t)ic * HW + gy * IMG + gx];
    in_s[ic][yy][xx] = v;
  }
  __syncthreads();
  const int y = tid >> 4, xc = tid & 15;
  float* outb = out + ((size_t)b * C4 + g * 8) * HW + (ty0 + y) * IMG + tx0 + xc;
  const bool isL = (g >= 144);           // uniform per block
  for (int oc = 0; oc < 8; ++oc) {
    float acc = 0.f;
    const float* wp = &ws_[oc * 72];
#pragma unroll
    for (int ic = 0; ic < 8; ++ic)
#pragma unroll
      for (int ky = 0; ky < 3; ++ky)
#pragma unroll
        for (int kx = 0; kx < 3; ++kx)
          acc += wp[ic * 9 + ky * 3 + kx] * in_s[ic][y + ky][xc + kx];
    outb[(size_t)oc * HW] = acc;
    if (isL) {
      red[tid] = acc;
      __syncthreads();
      for (int s = 128; s > 0; s >>= 1) {
        if (tid < s) red[tid] += red[tid + s];
        __syncthreads();
      }
      if (tid == 0) atomicAdd(&lsum[b * DIM1 + (g - 144) * 8 + oc], red[0]);
      __syncthreads();
    }
  }
}

// =====================================================================
// K3: channel-attention MLP: avg -> fc1/relu -> fc2 -> y*sigmoid(y)
// grid 4, block 384
// =====================================================================
__global__ __launch_bounds__(384) void k3_mlp(const float* __restrict__ lsum,
                                              const float* __restrict__ fc1w,
                                              const float* __restrict__ fc1b,
                                              const float* __restrict__ fc2w,
                                              const float* __restrict__ fc2b,
                                              float* __restrict__ L) {
  const int b = blockIdx.x, t = threadIdx.x;
  __shared__ float avg[DIM1], h1[24];
  avg[t] = lsum[b * DIM1 + t] * (1.f / (float)HW);
  __syncthreads();
  if (t < 24) {
    float s = fc1b[t];
    for (int c = 0; c < DIM1; ++c) s += avg[c] * fc1w[t * DIM1 + c];
    h1[t] = fmaxf(s, 0.f);
  }
  __syncthreads();
  float s = fc2b[t];
  for (int j = 0; j < 24; ++j) s += h1[j] * fc2w[t * 24 + j];
  L[b * DIM1 + t] = s / (1.f + __expf(-s));   // s * sigmoid(s)
}

// =====================================================================
// K4a: per (b,head): S = qhat khat^T * temp, softmax. Norms fused into
// the K-chunk loop (single pass over q,k). Staging uses async-to-LDS
// (B128, ASYNCcnt) when the builtin exists. grid 32, block 288 (9 waves
// -> 3x3 tile grid of the 48x48 output). Stride 132: 16B-aligned rows
// AND conflict-free column reads (banks 4r+c vs 4r+c+2 are disjoint).
// =====================================================================
#define ST4A 132
__global__ __launch_bounds__(288) void k4a_attn(const float* __restrict__ qkv2,
                                                const float* __restrict__ temp,
                                                float* __restrict__ attn) {
  const int bh = blockIdx.x;
  const int b = bh >> 3, h = bh & 7;
  const int tid = threadIdx.x;
  const int wave = tid >> 5, lane = tid & 31;
  const float* qbase = qkv2 + ((size_t)b * C4 + h * HC) * HW;
  const float* kbase = qkv2 + ((size_t)b * C4 + DIM1 + h * HC) * HW;
  __shared__ float inv_[96];
  __shared__ __align__(16) float qs[48 * ST4A];
  __shared__ __align__(16) float ks[48 * ST4A];
  __shared__ float sm[48 * 48];

  const int mt = wave / 3, nt = wave % 3;
  const int lr = lane & 15, hi = lane >> 4;
  v8f acc = {};
  float ss[11];                       // per-lane sumsq partials, rows wave+9j
#pragma unroll
  for (int j = 0; j < 11; ++j) ss[j] = 0.f;

  for (int ch = 0; ch < HW; ch += 128) {
    __syncthreads();
#ifdef HAVE_ASYNC_LDS
    // 2 arrays * 48 rows * 32 x 16B segments = 3072 async B128 transfers
    for (int i = tid; i < 3072; i += 288) {
      const int arr = (i >= 1536) ? 1 : 0;
      const int rem = i - arr * 1536;
      const int r = rem >> 5, seg = rem & 31;
      const float* gsrc = (arr ? kbase : qbase) + (size_t)r * HW + ch + seg * 4;
      float* ldst = (arr ? ks : qs) + r * ST4A + seg * 4;
      __builtin_amdgcn_global_load_async_to_lds_b128(
          (AS1 v4i*)(unsigned long long)(size_t)gsrc,
          (AS3 v4i*)(unsigned)(size_t)ldst, 0, 0);
    }
    __builtin_amdgcn_s_wait_asynccnt(0);
#else
    for (int i = tid; i < 48 * 128; i += 288) {
      int r = i >> 7, c = i & 127;
      qs[r * ST4A + c] = qbase[(size_t)r * HW + ch + c];
      ks[r * ST4A + c] = kbase[(size_t)r * HW + ch + c];
    }
#endif
    __syncthreads();
    // WMMA accumulation over this chunk
    for (int kk = 0; kk < 128; kk += 4) {
      const int ka = kk + hi * 2;
      v2f a, bb;
      a.x  = qs[(mt * 16 + lr) * ST4A + ka];
      a.y  = qs[(mt * 16 + lr) * ST4A + ka + 1];
      bb.x = ks[(nt * 16 + lr) * ST4A + ka];
      bb.y = ks[(nt * 16 + lr) * ST4A + ka + 1];
      acc = __builtin_amdgcn_wmma_f32_16x16x4_f32(false, a, false, bb,
                                                  (short)0, acc, false, false);
    }
    // fused sum-of-squares for this chunk (wave-uniform row selection)
#pragma unroll
    for (int j = 0; j < 11; ++j) {
      const int row = wave + 9 * j;
      if (row < 96) {
        const float* src = (row < 48) ? &qs[row * ST4A] : &ks[(row - 48) * ST4A];
#pragma unroll
        for (int t = 0; t < 4; ++t) {
          float v = src[lane * 4 + t];
          ss[j] += v * v;
        }
      }
    }
  }
  // finalize inverse norms: 1/max(||row||, 1e-12)
#pragma unroll
  for (int j = 0; j < 11; ++j) {
    const int row = wave + 9 * j;
    if (row < 96) {
      float s = ss[j];
      for (int off = 16; off > 0; off >>= 1) s += __shfl_xor(s, off, 32);
      if (lane == 0) inv_[row] = 1.f / fmaxf(sqrtf(s), 1e-12f);
    }
  }
  __syncthreads();
  const float tp = temp[h];
#pragma unroll
  for (int r = 0; r < 8; ++r) {
    int c = mt * 16 + r + 8 * hi;
    int d = nt * 16 + lr;
    sm[c * 48 + d] = acc[r] * inv_[c] * inv_[48 + d] * tp;
  }
  __syncthreads();
  // row softmax, write out
  if (tid < 48) {
    float mx = -1e30f;
    for (int d = 0; d < 48; ++d) mx = fmaxf(mx, sm[tid * 48 + d]);
    float sum = 0.f;
    for (int d = 0; d < 48; ++d) {
      float e = __expf(sm[tid * 48 + d] - mx);
      sm[tid * 48 + d] = e; sum += e;
    }
    float is = 1.f / sum;
    float* dst = attn + ((size_t)bh * 48 + tid) * 48;
    for (int d = 0; d < 48; ++d) dst[d] = sm[tid * 48 + d] * is;
  }
}

// =====================================================================
// K4b: out[b, h*48+c, n] = L[b, h*48+c] + sum_d attn[c,d] * v[d,n]
// grid (32, 64), block 256 (8 waves); attn cached in LDS (stride 49).
// =====================================================================
__global__ __launch_bounds__(256) void k4b_av(const float* __restrict__ qkv2,
                                              const float* __restrict__ attn,
                                              const float* __restrict__ L,
                                              float* __restrict__ out) {
  const int bh = blockIdx.x, b = bh >> 3, h = bh & 7;
  const int n0 = blockIdx.y * 256;
  const int tid = threadIdx.x, wave = tid >> 5, lane = tid & 31;
  __shared__ float As[48 * 49];
  for (int i = tid; i < 48 * 48; i += 256) {
    int r = i / 48, c = i % 48;
    As[r * 49 + c] = attn[(size_t)bh * 2304 + i];
  }
  __syncthreads();
  const float* vbase = qkv2 + ((size_t)b * C4 + 2 * DIM1 + h * HC) * HW;
  const int lr = lane & 15, hi = lane >> 4;
  for (int t = wave; t < 48; t += 8) {        // 3 mtiles x 16 ntiles
    const int mt = t / 16, p0 = n0 + (t % 16) * 16;
    v8f acc = {};
    for (int kk = 0; kk < 48; kk += 4) {
      const int ka = kk + hi * 2;
      v2f a, bb;
      a.x  = As[(mt * 16 + lr) * 49 + ka];
      a.y  = As[(mt * 16 + lr) * 49 + ka + 1];
      bb.x = vbase[(size_t)ka * HW + p0 + lr];
      bb.y = vbase[(size_t)(ka + 1) * HW + p0 + lr];
      acc = __builtin_amdgcn_wmma_f32_16x16x4_f32(false, a, false, bb,
                                                  (short)0, acc, false, false);
    }
#pragma unroll
    for (int r = 0; r < 8; ++r) {
      int c = mt * 16 + r + 8 * hi;
      out[((size_t)b * DIM1 + h * HC + c) * HW + p0 + lr] =
          acc[r] + L[b * DIM1 + h * HC + c];
    }
  }
}

// =====================================================================
// K5: out[b,o,p] = x[b,o,p] + sum_c po1_w[o,c] * ain[b,c,p]
// grid (128, 3, 4), block 256 (8 waves, one 16x16 tile each), K=384.
// Prefetch (global_prefetch_b8) 16 K-rows ahead on the B stream.
// =====================================================================
__global__ __launch_bounds__(256) void k5_proj(const float* __restrict__ ain,
                                               const float* __restrict__ w,
                                               const float* __restrict__ x,
                                               float* __restrict__ out) {
  const int b = blockIdx.z, m0 = blockIdx.y * 16;
  const int tid = threadIdx.x, wave = tid >> 5, lane = tid & 31;
  __shared__ float As[16 * 389];
  for (int i = tid; i < 16 * 384; i += 256) {
    int r = i / 384, c = i % 384;
    As[r * 389 + c] = w[(m0 + r) * 384 + c];
  }
  __syncthreads();
  const float* ab = ain + (size_t)b * DIM1 * HW;
  const int lr = lane & 15, hi = lane >> 4;
  const int p0 = blockIdx.x * 128 + wave * 16;
  v8f acc = {};
  for (int kk = 0; kk < 384; kk += 4) {
    const int ka = kk + hi * 2;
    if ((kk & 15) == 0 && kk + 16 < 384)
      __builtin_prefetch(&ab[(size_t)(ka + 16) * HW + p0 + lr], 0, 1);
    v2f a, bb;
    a.x  = As[lr * 389 + ka];
    a.y  = As[lr * 389 + ka + 1];
    bb.x = ab[(size_t)ka * HW + p0 + lr];
    bb.y = ab[(size_t)(ka + 1) * HW + p0 + lr];
    acc = __builtin_amdgcn_wmma_f32_16x16x4_f32(false, a, false, bb,
                                                (short)0, acc, false, false);
  }
#pragma unroll
  for (int r = 0; r < 8; ++r) {
    const int o = m0 + r + 8 * hi;
    const size_t idx = ((size_t)b * CIN + o) * HW + p0 + lr;
    out[idx] = acc[r] + x[idx];
  }
}

// =====================================================================
extern "C" void kernel_launch(void* const* d_in, const int* in_sizes, int n_in,
                              void* d_out, int out_size, void* d_ws, size_t ws_size,
                              hipStream_t stream) {
  const float* x     = (const float*)d_in[0];
  const float* qkvw  = (const float*)d_in[1];
  const float* dww   = (const float*)d_in[2];
  const float* po1w  = (const float*)d_in[3];
  const float* fc1w  = (const float*)d_in[4];
  const float* fc1b  = (const float*)d_in[5];
  const float* fc2w  = (const float*)d_in[6];
  const float* fc2b  = (const float*)d_in[7];
  const float* temp  = (const float*)d_in[8];
  float* out = (float*)d_out;

  const size_t QKV_BYTES = (size_t)B4 * C4 * HW * sizeof(float); // 402.6 MB
  char* base = (char*)d_ws;
  float* p_qkv    = (float*)(base);                       // k1 out, k2 in; reused as attnout by k4b
  float* p_qkv2   = (float*)(base + QKV_BYTES);           // k2 out (q,k,v,l)
  float* p_lsum   = (float*)(base + 2 * QKV_BYTES);       // 1536 floats
  float* p_L      = p_lsum + B4 * DIM1;                   // 1536 floats
  float* p_attn   = p_L + B4 * DIM1;                      // 4*8*48*48 floats
  float* p_aout   = p_qkv;                                // reuse qkv region

  (void)hipMemsetAsync(p_lsum, 0, B4 * DIM1 * sizeof(float), stream);

  k1_qkv <<<dim3(32, 96, 4),  256, 0, stream>>>(x, qkvw, p_qkv);
  k2_gconv<<<dim3(64, 192, 4), 256, 0, stream>>>(p_qkv, dww, p_qkv2, p_lsum);
  k3_mlp  <<<dim3(4),          384, 0, stream>>>(p_lsum, fc1w, fc1b, fc2w, fc2b, p_L);
  k4a_attn<<<dim3(32),         288, 0, stream>>>(p_qkv2, temp, p_attn);
  k4b_av  <<<dim3(32, 64),     256, 0, stream>>>(p_qkv2, p_attn, p_L, p_aout);
  k5_proj <<<dim3(128, 3, 4),  256, 0, stream>>>(p_aout, po1w, x, out);
}